// GRU_75316546503276
// MI455X (gfx1250) — compile-verified
//
#include <hip/hip_runtime.h>

// Problem dims (match the JAX reference)
#define Bsz   256
#define Tsz   512
#define DIN   32
#define Hsz   256
#define LD1   512
#define LD2   256
#define DOUT  12

// LDS row pitch (floats): 256 + 4 so a wave's A-fragment read
// (lane m -> row m, byte offset 4*(m*KP + kk)) hits banks 4m+kk..4m+kk+3,
// i.e. all 64 banks exactly once -> conflict-free ds_load_b64.
#define KP    260

typedef float v2f __attribute__((ext_vector_type(2)));
typedef float v8f __attribute__((ext_vector_type(8)));

// ---------------------------------------------------------------------------
// Fused 3-gate accumulation for one 16x16 output tile.
// A fragment is read ONCE per k-chunk from LDS and feeds 3 WMMAs (r,z,n).
//   shA : 16 x KP LDS tile (batch rows of the activation)
//   W   : row-major (3H x K) weight; gate g rows start at g*Hsz
// A VGPR layout (ISA 7.12.2, 32-bit A 16x4): lane m<16 holds M=m with K,K+1;
// lanes 16..31 hold K+2,K+3. B loaded symmetrically (N = lane&15).
// ---------------------------------------------------------------------------
__device__ __forceinline__ void gates_accum(
    const float* __restrict__ shA,
    const float* __restrict__ W, int ldw, int col0, int K,
    v8f& gR, v8f& gZ, v8f& gN)
{
    const int lane  = threadIdx.x & 31;
    const int mn    = lane & 15;
    const int khalf = lane >> 4;              // 0 or 1
    const float* aRow = shA + mn * KP;
    const float* wR = W + (col0 + mn) * ldw;  // r-gate rows
    const float* wZ = wR + Hsz * ldw;         // z-gate rows
    const float* wN = wZ + Hsz * ldw;         // n-gate rows
    for (int k = 0; k < K; k += 4) {
        const int kk = k + 2 * khalf;
        v2f a, br, bz, bn;
        a[0]  = aRow[kk]; a[1]  = aRow[kk + 1];
        br[0] = wR[kk];   br[1] = wR[kk + 1];
        bz[0] = wZ[kk];   bz[1] = wZ[kk + 1];
        bn[0] = wN[kk];   bn[1] = wN[kk + 1];
        gR = __builtin_amdgcn_wmma_f32_16x16x4_f32(false, a, false, br, (short)0, gR, false, false);
        gZ = __builtin_amdgcn_wmma_f32_16x16x4_f32(false, a, false, bz, (short)0, gZ, false, false);
        gN = __builtin_amdgcn_wmma_f32_16x16x4_f32(false, a, false, bn, (short)0, gN, false, false);
    }
}

__device__ __forceinline__ float sigmoidf_fast(float x)
{
    return 1.0f / (1.0f + __expf(-x));
}

// ---------------------------------------------------------------------------
// One GRU cell step. Block = 128 threads (4 waves) covering 16 batch rows x
// 64 hidden cols; each wave owns one 16x16 tile. X and Hin row-tiles are
// staged in LDS once per block and shared by all 4 waves / 3 gates.
//   X   : (Bsz x Dx) timestep input (ldx = T*DIN for layer0, Hsz for layer1)
//   Hin : (Bsz x Hsz) previous hidden state (read-only this step)
//   Wih : (3H x Dx), Whh : (3H x Hsz), gate order [r | z | n]
// n-gate input/hidden halves kept separate (r scales only hn + bhh_n).
// ---------------------------------------------------------------------------
__global__ __launch_bounds__(128) void gru_cell_kernel(
    const float* __restrict__ X, int ldx, int Dx,
    const float* __restrict__ Hin,
    const float* __restrict__ Wih, const float* __restrict__ Whh,
    const float* __restrict__ bih, const float* __restrict__ bhh,
    float* __restrict__ Hout)
{
    __shared__ float shX[16 * KP];
    __shared__ float shH[16 * KP];

    const int row0 = blockIdx.x * 16;     // batch tile
    const int tid  = threadIdx.x;

    // Cooperative stage of the activation tiles (coalesced along columns)
    for (int i = tid; i < 16 * Dx; i += 128) {
        const int r = i / Dx, c = i % Dx;
        shX[r * KP + c] = X[(row0 + r) * ldx + c];
    }
    for (int i = tid; i < 16 * Hsz; i += 128) {
        const int r = i >> 8, c = i & (Hsz - 1);
        shH[r * KP + c] = Hin[(row0 + r) * Hsz + c];
    }
    __syncthreads();

    const int wave = tid >> 5;
    const int col0 = blockIdx.y * 64 + wave * 16;   // hidden tile of this wave

    v8f accR  = {};
    v8f accZ  = {};
    v8f accXN = {};
    v8f accHN = {};

    gates_accum(shX, Wih, Dx,  col0, Dx,  accR, accZ, accXN);  // x @ Wih^T
    gates_accum(shH, Whh, Hsz, col0, Hsz, accR, accZ, accHN);  // h @ Whh^T

    const int lane  = tid & 31;
    const int n     = lane & 15;
    const int khalf = lane >> 4;
    const int c     = col0 + n;

    const float br  = bih[c]           + bhh[c];
    const float bz  = bih[Hsz + c]     + bhh[Hsz + c];
    const float bxn = bih[2 * Hsz + c];
    const float bhn = bhh[2 * Hsz + c];

#pragma unroll
    for (int i = 0; i < 8; ++i) {
        const int ml = i + khalf * 8;                 // row within tile
        const float r     = sigmoidf_fast(accR[i] + br);
        const float z     = sigmoidf_fast(accZ[i] + bz);
        const float hn    = accHN[i] + bhn;
        const float ng    = tanhf(accXN[i] + bxn + r * hn);
        const float hprev = shH[ml * KP + c];         // Hin tile already in LDS
        Hout[(row0 + ml) * Hsz + c] = (1.0f - z) * ng + z * hprev;
    }
}

// ---------------------------------------------------------------------------
// Dense layer Out = act(A @ W^T + b), one 16x16 WMMA tile per wave
// (decoder FC1/FC2; runs twice total so kept simple, no LDS staging).
// ---------------------------------------------------------------------------
__global__ __launch_bounds__(32) void fc_kernel(
    const float* __restrict__ A, int K,
    const float* __restrict__ W, const float* __restrict__ bias,
    float* __restrict__ Out, int N, int do_relu)
{
    const int row0 = blockIdx.x * 16;
    const int col0 = blockIdx.y * 16;

    const int lane  = threadIdx.x & 31;
    const int mn    = lane & 15;
    const int khalf = lane >> 4;
    const float* aRow = A + (row0 + mn) * K;
    const float* wRow = W + (col0 + mn) * K;

    v8f acc = {};
    for (int k = 0; k < K; k += 4) {
        const int kk = k + 2 * khalf;
        v2f a, b;
        a[0] = aRow[kk]; a[1] = aRow[kk + 1];
        b[0] = wRow[kk]; b[1] = wRow[kk + 1];
        acc = __builtin_amdgcn_wmma_f32_16x16x4_f32(false, a, false, b, (short)0, acc, false, false);
    }

    const float b = bias[col0 + mn];
#pragma unroll
    for (int i = 0; i < 8; ++i) {
        const int m = row0 + i + khalf * 8;
        float v = acc[i] + b;
        if (do_relu) v = fmaxf(v, 0.0f);
        Out[m * N + col0 + mn] = v;
    }
}

// Final (B x 12) layer — N=12 is not a 16-tile, plain VALU dot kernel.
__global__ void fc_out_kernel(const float* __restrict__ A,
                              const float* __restrict__ W3,
                              const float* __restrict__ b3,
                              float* __restrict__ Out)
{
    const int idx = blockIdx.x * blockDim.x + threadIdx.x;
    if (idx >= Bsz * DOUT) return;
    const int b = idx / DOUT;
    const int o = idx % DOUT;
    const float* a = A + b * LD2;
    const float* w = W3 + o * LD2;
    float s = b3[o];
    for (int k = 0; k < LD2; ++k) s = fmaf(a[k], w[k], s);
    Out[idx] = s;
}

__global__ void zero_kernel(float* __restrict__ p, int n)
{
    const int idx = blockIdx.x * blockDim.x + threadIdx.x;
    if (idx < n) p[idx] = 0.0f;
}

// ---------------------------------------------------------------------------
extern "C" void kernel_launch(void* const* d_in, const int* in_sizes, int n_in,
                              void* d_out, int out_size, void* d_ws, size_t ws_size,
                              hipStream_t stream)
{
    const float* x    = (const float*)d_in[0];
    const float* Wih0 = (const float*)d_in[1];
    const float* Whh0 = (const float*)d_in[2];
    const float* bih0 = (const float*)d_in[3];
    const float* bhh0 = (const float*)d_in[4];
    const float* Wih1 = (const float*)d_in[5];
    const float* Whh1 = (const float*)d_in[6];
    const float* bih1 = (const float*)d_in[7];
    const float* bhh1 = (const float*)d_in[8];
    const float* W1   = (const float*)d_in[9];
    const float* b1   = (const float*)d_in[10];
    const float* W2   = (const float*)d_in[11];
    const float* b2   = (const float*)d_in[12];
    const float* W3   = (const float*)d_in[13];
    const float* b3   = (const float*)d_in[14];
    // d_in[15] = init_memory scalar (initial state is zeros; unused)

    float* ws   = (float*)d_ws;
    float* h0a  = ws;
    float* h0b  = h0a + Bsz * Hsz;
    float* h1a  = h0b + Bsz * Hsz;
    float* h1b  = h1a + Bsz * Hsz;
    float* dec1 = h1b + Bsz * Hsz;          // (B x LD1)
    float* dec2 = dec1 + Bsz * LD1;         // (B x LD2)

    // h0 = h1 = 0 at t=0 (ws is poisoned; must init every call)
    {
        const int zn = 4 * Bsz * Hsz;
        zero_kernel<<<(zn + 255) / 256, 256, 0, stream>>>(ws, zn);
    }

    float* h0i = h0a; float* h0o = h0b;
    float* h1i = h1a; float* h1o = h1b;

    const dim3 grid(Bsz / 16, Hsz / 64);    // 16 x 4 blocks, 4 waves each
    for (int t = 0; t < Tsz; ++t) {
        // layer 0: x_t lives inside x[B,T,DIN] -> base x + t*DIN, lda = T*DIN
        gru_cell_kernel<<<grid, 128, 0, stream>>>(
            x + (size_t)t * DIN, Tsz * DIN, DIN,
            h0i, Wih0, Whh0, bih0, bhh0, h0o);
        // layer 1 consumes layer 0's fresh hidden state of this same step
        gru_cell_kernel<<<grid, 128, 0, stream>>>(
            h0o, Hsz, Hsz,
            h1i, Wih1, Whh1, bih1, bhh1, h1o);
        float* t0 = h0i; h0i = h0o; h0o = t0;
        float* t1 = h1i; h1i = h1o; h1o = t1;
    }
    // after the final swap, h1i holds y1[:, T-1, :]
    fc_kernel<<<dim3(Bsz / 16, LD1 / 16), 32, 0, stream>>>(h1i, Hsz, W1, b1, dec1, LD1, 1);
    fc_kernel<<<dim3(Bsz / 16, LD2 / 16), 32, 0, stream>>>(dec1, LD1, W2, b2, dec2, LD2, 1);
    fc_out_kernel<<<(Bsz * DOUT + 255) / 256, 256, 0, stream>>>(dec2, W3, b3, (float*)d_out);
}